// DOMINO_85598698209587
// MI455X (gfx1250) — compile-verified
//
#include <hip/hip_runtime.h>

// DOMINO fused loss (CE + generalized-dice + penalty) for MI455X / gfx1250.
//
// Everything reduces to, per image n:
//   S_n[m][c] = sum over voxels with target==m of softmax(x_v)[c]   (12x12)
// plus a scalar CE sum. S_n is accumulated with V_WMMA_F32_16X16X4_F32
// (one-hot(t) outer-product softmax rows, 4 voxels per WMMA, fp32-exact).
// Bandwidth-bound: ~218 MB single pass -> ~9.5 us floor at 23.3 TB/s.
//
// Workspace: 2 * GX * 257 floats = 526,336 bytes of d_ws.

#define HWZ   (128*128*128)      // spatial voxels per image
#define CCH   12                 // channels
#define GX    256                // blocks per image
#define TPB   256                // threads per block (8 wave32)
#define ITERS (HWZ / (GX * TPB)) // 32, exact
#define SMOOTHV 1e-05
#define BETAV   3.0

typedef __attribute__((ext_vector_type(2))) float v2f;
typedef __attribute__((ext_vector_type(4))) float v4f;
typedef __attribute__((ext_vector_type(8))) float v8f;

__global__ __launch_bounds__(TPB) void domino_main_kernel(
    const float* __restrict__ inp,   // (2,12,HWZ)
    const int*   __restrict__ tgt,   // (2,HWZ)
    float*       __restrict__ ws)    // (2,GX,257)
{
  // Per-wave private staging: 32 voxels x 16 floats (p, zero-padded) + 32 targets.
  __shared__ float ldsP[8][32 * 16];   // 16 KB
  __shared__ int   ldsT[8][32];        //  1 KB
  __shared__ float ldsS[8][256];       //  8 KB  (per-wave 16x16 S tiles)
  __shared__ float ldsCE[TPB];         //  1 KB

  const int tid  = threadIdx.x;
  const int lane = tid & 31;
  const int wave = tid >> 5;
  const int hi   = lane >> 4;   // half-wave selector (K group / row group)
  const int nn   = lane & 15;   // column / row index 0..15
  const int n    = blockIdx.y;

  const float* __restrict__ xin = inp + (size_t)n * CCH * HWZ;
  const int*   __restrict__ tg  = tgt + (size_t)n * HWZ;
  const unsigned gbase = blockIdx.x * TPB + (unsigned)tid;

  float* pw = &ldsP[wave][0];
  int*   tw = &ldsT[wave][0];

  v8f   acc = {0.f,0.f,0.f,0.f,0.f,0.f,0.f,0.f};  // S tile accumulator (16x16 f32)
  float ce  = 0.f;

  for (int it = 0; it < ITERS; ++it) {
    const unsigned v = (unsigned)it * (GX * TPB) + gbase;
    const int t = tg[v];
    if (it + 1 < ITERS) {                        // wave-uniform branch, EXEC stays full
      __builtin_prefetch(&tg[v + GX * TPB], 0, 0);
      __builtin_prefetch(&xin[v + GX * TPB], 0, 0);
    }

    // ---- streaming softmax over 12 channels (coalesced per channel) ----
    float x[CCH];
    #pragma unroll
    for (int c = 0; c < CCH; ++c) x[c] = xin[(size_t)c * HWZ + v];
    float mx = x[0];
    #pragma unroll
    for (int c = 1; c < CCH; ++c) mx = fmaxf(mx, x[c]);
    float s = 0.f, xt = 0.f;
    float e[CCH];
    #pragma unroll
    for (int c = 0; c < CCH; ++c) {
      e[c] = __expf(x[c] - mx);
      s += e[c];
      xt = (c == t) ? x[c] : xt;                 // v_cndmask, no divergence
    }
    const float inv = 1.0f / s;
    ce += (mx + __logf(s)) - xt;                 // -log p_t

    // ---- stage normalized p (padded to 16) and t into this wave's LDS strip ----
    v4f q0 = { e[0]*inv, e[1]*inv, e[2]*inv,  e[3]*inv  };
    v4f q1 = { e[4]*inv, e[5]*inv, e[6]*inv,  e[7]*inv  };
    v4f q2 = { e[8]*inv, e[9]*inv, e[10]*inv, e[11]*inv };
    v4f q3 = { 0.f, 0.f, 0.f, 0.f };
    *(v4f*)(pw + lane * 16 + 0)  = q0;
    *(v4f*)(pw + lane * 16 + 4)  = q1;
    *(v4f*)(pw + lane * 16 + 8)  = q2;
    *(v4f*)(pw + lane * 16 + 12) = q3;
    tw[lane] = t;
    asm volatile("" ::: "memory");               // keep compiler from reordering LDS ld/st
    // (same-wave DS ops are architecturally in-order; no barrier needed)

    // ---- 8 x WMMA: S += OneHot(t[4i..4i+3])^T x P[4i..4i+3] ----
    // A 16x4 layout (documented): lanes 0-15 hold K=0/1 in VGPR0/1, lanes 16-31 K=2/3.
    // B 4x16 layout (mirror of A): VGPR0 = rows K=0 (lanes 0-15) / K=2 (lanes 16-31),
    //                              VGPR1 = rows K=1 / K=3.
    #pragma unroll
    for (int i = 0; i < 8; ++i) {
      const int k0 = i * 4 + hi * 2;
      v2f B, A;
      B.x = pw[k0 * 16 + nn];
      B.y = pw[(k0 + 1) * 16 + nn];
      const int t0 = tw[k0];
      const int t1 = tw[k0 + 1];
      A.x = (t0 == nn) ? 1.0f : 0.0f;
      A.y = (t1 == nn) ? 1.0f : 0.0f;
      acc = __builtin_amdgcn_wmma_f32_16x16x4_f32(
          /*neg_a=*/false, A, /*neg_b=*/false, B,
          /*c_mod=*/(short)0, acc, /*reuse_a=*/false, /*reuse_b=*/false);
    }
  }

  // ---- spill wave S tile (documented C/D layout: VGPR r -> row r + hi*8, col nn) ----
  #pragma unroll
  for (int r = 0; r < 8; ++r)
    ldsS[wave][(r + hi * 8) * 16 + nn] = acc[r];
  ldsCE[tid] = ce;
  __syncthreads();

  // ---- deterministic block combine (fixed summation order) ----
  float* wsb = ws + ((size_t)n * GX + blockIdx.x) * 257;
  float ssum = 0.f;
  #pragma unroll
  for (int w = 0; w < 8; ++w) ssum += ldsS[w][tid];
  wsb[tid] = ssum;
  if (tid == 0) {
    float c2 = 0.f;
    for (int j = 0; j < TPB; ++j) c2 += ldsCE[j];
    wsb[256] = c2;
  }
}

__global__ __launch_bounds__(256) void domino_final_kernel(
    const float* __restrict__ Mp,    // (12,12) matrix_penalty
    const float* __restrict__ ws,    // (2,GX,257)
    float*       __restrict__ out)
{
  __shared__ double S[2][256];
  __shared__ double ceS;
  const int tid = threadIdx.x;

  for (int n = 0; n < 2; ++n) {
    double a = 0.0;
    for (int b = 0; b < GX; ++b)
      a += (double)ws[((size_t)n * GX + b) * 257 + tid];
    S[n][tid] = a;
  }
  if (tid == 0) {
    double c = 0.0;
    for (int n = 0; n < 2; ++n)
      for (int b = 0; b < GX; ++b)
        c += (double)ws[((size_t)n * GX + b) * 257 + 256];
    ceS = c;
  }
  __syncthreads();

  if (tid == 0) {
    const double NV = 2.0 * (double)HWZ;         // voxels total
    const double ce = ceS / NV;
    double dice = 0.0, pen = 0.0;
    for (int n = 0; n < 2; ++n) {
      for (int c = 0; c < CCH; ++c) {
        const double inter = S[n][c * 16 + c];
        double go = 0.0, po = 0.0;
        for (int c2 = 0; c2 < CCH; ++c2) {
          go += S[n][c * 16 + c2];   // row sum  = ground_o (count, since sum_c p = 1)
          po += S[n][c2 * 16 + c];   // col sum  = pred_o
        }
        dice += 1.0 - (2.0 * inter + SMOOTHV) / (go + po + SMOOTHV);
      }
      for (int r = 0; r < CCH; ++r)
        for (int c = 0; c < CCH; ++c)
          pen += (double)Mp[r * CCH + c] * S[n][r * 16 + c];
    }
    dice /= 24.0;                                // mean over (N,C)
    pen = BETAV * pen / NV;                      // mean over voxels * beta
    out[0] = (float)(ce + dice + pen);
  }
}

extern "C" void kernel_launch(void* const* d_in, const int* in_sizes, int n_in,
                              void* d_out, int out_size, void* d_ws, size_t ws_size,
                              hipStream_t stream) {
  const float* inp = (const float*)d_in[0];          // (2,12,128,128,128) fp32
  const int*   tgt = (const int*)  d_in[1];          // (2,1,128,128,128) int
  const float* Mp  = (const float*)d_in[2];          // (12,12) fp32
  float* ws  = (float*)d_ws;                         // needs 526,336 bytes
  float* out = (float*)d_out;

  dim3 grid(GX, 2);
  domino_main_kernel<<<grid, TPB, 0, stream>>>(inp, tgt, ws);
  domino_final_kernel<<<1, 256, 0, stream>>>(Mp, ws, out);
}